// BipartiteGNN_19920058318953
// MI455X (gfx1250) — compile-verified
//
#include <hip/hip_runtime.h>

typedef __attribute__((ext_vector_type(2))) float v2f;
typedef __attribute__((ext_vector_type(8))) float v8f;

// ---------------------------------------------------------------------------
// Zero-fill (graph-capture-safe replacement for hipMemsetAsync)
// ---------------------------------------------------------------------------
__global__ void fill_zero_kernel(float4* __restrict__ p, long long n4) {
  long long i = (long long)blockIdx.x * blockDim.x + threadIdx.x;
  if (i < n4) p[i] = make_float4(0.f, 0.f, 0.f, 0.f);
}

// ---------------------------------------------------------------------------
// Edge scatter-add: agg[dst] += x_src[src]   (aggr='add' SAGE message passing)
// 16 threads per edge, float4 per thread -> 256B contiguous per edge,
// native global_atomic_add_f32 via unsafeAtomicAdd (L2-near-memory atomics).
// ---------------------------------------------------------------------------
__global__ void scatter_add_kernel(const float* __restrict__ xsrc,
                                   const int* __restrict__ edges,
                                   float* __restrict__ agg, int E) {
  long long tid = (long long)blockIdx.x * blockDim.x + threadIdx.x;
  if (tid >= (long long)E * 16) return;
  int e = (int)(tid >> 4);
  int c = ((int)tid & 15) << 2;
  int s = edges[e];          // edges[0][e]
  int d = edges[E + e];      // edges[1][e]
  const float4 v = *(const float4*)(xsrc + (long long)s * 64 + c);
  float* dst = agg + (long long)d * 64 + c;
  unsafeAtomicAdd(dst + 0, v.x);
  unsafeAtomicAdd(dst + 1, v.y);
  unsafeAtomicAdd(dst + 2, v.z);
  unsafeAtomicAdd(dst + 3, v.w);
}

// ---------------------------------------------------------------------------
// out = agg @ Wl + x_dst @ Wr + b   using V_WMMA_F32_16X16X4_F32 (fp32 WMMA).
// Block: 256 threads = 8 waves. Wave w computes rows [blk*128 + w*16, +16),
// all 64 output columns as 4 accumulator tiles of 16x16.
//
// Weights are staged into LDS PRE-SWIZZLED into WMMA B-fragment layout:
//   sW[k2*64 + n] = (float2){ W[2*k2][n], W[2*k2+1][n] }
// so each B fragment is one aligned ds_load_b64 into an even VGPR pair
// (no v_mov repacking).
//
// Fragment layout per CDNA5 ISA 7.12.2:
//  A (16x4 f32): lane (half,lm): VGPR0=A[lm][k0+2*half], VGPR1=A[lm][k0+2*half+1]
//  B (4x16 f32): lane (half,lm): VGPR0=B[k0+2*half][n],  VGPR1=B[k0+2*half+1][n]
//  C/D (16x16):  VGPR r: lanes0-15 -> (M=r, N=lm); lanes16-31 -> (M=r+8, N=lm)
// ---------------------------------------------------------------------------
__global__ __launch_bounds__(256)
void sage_gemm_kernel(const float* __restrict__ agg,
                      const float* __restrict__ xdst,
                      const float* __restrict__ Wl,
                      const float* __restrict__ Wr,
                      const float* __restrict__ bias,
                      float* __restrict__ out, int n) {
  __shared__ v2f sWl[32 * 64];   // 16 KB, fragment-swizzled
  __shared__ v2f sWr[32 * 64];   // 16 KB, fragment-swizzled
  for (int idx = threadIdx.x; idx < 32 * 64; idx += 256) {
    const int k2 = idx >> 6;
    const int nc = idx & 63;
    sWl[idx] = (v2f){Wl[(2 * k2) * 64 + nc], Wl[(2 * k2 + 1) * 64 + nc]};
    sWr[idx] = (v2f){Wr[(2 * k2) * 64 + nc], Wr[(2 * k2 + 1) * 64 + nc]};
  }
  __syncthreads();

  const int wave = threadIdx.x >> 5;
  const int lane = threadIdx.x & 31;
  const int half = lane >> 4;   // 0 or 1
  const int lm   = lane & 15;   // row index (A) / col index (B)
  const int rowbase = blockIdx.x * 128 + wave * 16;
  if (rowbase >= n) return;     // wave-uniform: EXEC stays all-1s for WMMA

  v8f acc[4] = {};

  const float* __restrict__ arow = agg  + (long long)(rowbase + lm) * 64;
  const float* __restrict__ xrow = xdst + (long long)(rowbase + lm) * 64;

#pragma unroll
  for (int k0 = 0; k0 < 64; k0 += 4) {
    const int koff  = k0 + 2 * half;       // even -> 8B aligned
    const int k2row = (koff >> 1) << 6;    // (koff/2) * 64
    v2f aA = *(const v2f*)(arow + koff);   // global_load_b64
    v2f aX = *(const v2f*)(xrow + koff);
#pragma unroll
    for (int nt = 0; nt < 4; ++nt) {
      const int ncol = nt * 16 + lm;
      v2f bL = sWl[k2row + ncol];          // ds_load_b64, pre-swizzled pair
      v2f bR = sWr[k2row + ncol];
      acc[nt] = __builtin_amdgcn_wmma_f32_16x16x4_f32(
          false, aA, false, bL, (short)0, acc[nt], false, false);
      acc[nt] = __builtin_amdgcn_wmma_f32_16x16x4_f32(
          false, aX, false, bR, (short)0, acc[nt], false, false);
    }
  }

#pragma unroll
  for (int nt = 0; nt < 4; ++nt) {
    const int ncol = nt * 16 + lm;
    const float bv = bias[ncol];
#pragma unroll
    for (int r = 0; r < 8; ++r) {
      const int row = rowbase + r + 8 * half;
      out[(long long)row * 64 + ncol] = acc[nt][r] + bv;
    }
  }
}

// ---------------------------------------------------------------------------
// Column-wise global sum: pooled[c] += sum_rows x[row][c]
// 64 consecutive threads form a group striding over rows -> coalesced loads,
// one atomic per column per group.
// ---------------------------------------------------------------------------
__global__ void pool_kernel(const float* __restrict__ x,
                            float* __restrict__ pooled, int n) {
  int g = blockIdx.x * blockDim.x + threadIdx.x;
  int col = g & 63;
  int group = g >> 6;
  int ngroups = (gridDim.x * blockDim.x) >> 6;
  float sum = 0.f;
  for (int row = group; row < n; row += ngroups)
    sum += x[(long long)row * 64 + col];
  unsafeAtomicAdd(&pooled[col], sum);
}

// pooled[0:64] . W_lin + b_lin -> out[0]
__global__ void finish_kernel(const float* __restrict__ pooled,
                              const float* __restrict__ Wlin,
                              const float* __restrict__ blin,
                              float* __restrict__ out) {
  __shared__ float s[64];
  int t = threadIdx.x;
  s[t] = pooled[t] * Wlin[t];
  __syncthreads();
  for (int st = 32; st > 0; st >>= 1) {
    if (t < st) s[t] += s[t + st];
    __syncthreads();
  }
  if (t == 0) out[0] = s[0] + blin[0];
}

// ---------------------------------------------------------------------------
extern "C" void kernel_launch(void* const* d_in, const int* in_sizes, int n_in,
                              void* d_out, int out_size, void* d_ws, size_t ws_size,
                              hipStream_t stream) {
  const float* x_s    = (const float*)d_in[0];
  const float* x_t    = (const float*)d_in[1];
  const int*   e_s2t  = (const int*)d_in[2];
  const int*   e_t2s  = (const int*)d_in[3];
  const float* Wl_s2t = (const float*)d_in[4];
  const float* Wr_s2t = (const float*)d_in[5];
  const float* b_s2t  = (const float*)d_in[6];
  const float* Wl_t2s = (const float*)d_in[7];
  const float* Wr_t2s = (const float*)d_in[8];
  const float* b_t2s  = (const float*)d_in[9];
  const float* W_lin  = (const float*)d_in[10];
  const float* b_lin  = (const float*)d_in[11];
  float* out = (float*)d_out;

  const int NS = in_sizes[0] / 64;
  const int NT = in_sizes[1] / 64;
  const int E1 = in_sizes[2] / 2;
  const int E2 = in_sizes[3] / 2;
  const int L  = in_sizes[4] / (64 * 64);

  const size_t nodeFloats = (size_t)((NS > NT) ? NS : NT) * 64;
  float* ws     = (float*)d_ws;
  float* xs_a   = ws;
  float* xt_a   = xs_a + nodeFloats;
  float* xs_b   = xt_a + nodeFloats;
  float* xt_b   = xs_b + nodeFloats;
  float* agg    = xt_b + nodeFloats;
  float* pooled = agg + nodeFloats;  // 64 floats

  float* bufs_s[2] = {xs_a, xs_b};
  float* bufs_t[2] = {xt_a, xt_b};

  const float* xs_in = x_s;
  const float* xt_in = x_t;

  for (int l = 0; l < L; ++l) {
    float* xs_out = bufs_s[l & 1];
    float* xt_out = bufs_t[l & 1];

    // ---- s -> t direction: out_t = agg(xs over e_s2t) @ Wl + xt @ Wr + b
    {
      long long n4 = (long long)NT * 16;
      fill_zero_kernel<<<(int)((n4 + 255) / 256), 256, 0, stream>>>((float4*)agg, n4);
      long long th = (long long)E1 * 16;
      scatter_add_kernel<<<(int)((th + 255) / 256), 256, 0, stream>>>(xs_in, e_s2t, agg, E1);
      sage_gemm_kernel<<<(NT + 127) / 128, 256, 0, stream>>>(
          agg, xt_in, Wl_s2t + (size_t)l * 4096, Wr_s2t + (size_t)l * 4096,
          b_s2t + (size_t)l * 64, xt_out, NT);
    }
    // ---- t -> s direction: out_s = agg(xt over e_t2s) @ Wl + xs @ Wr + b
    {
      long long n4 = (long long)NS * 16;
      fill_zero_kernel<<<(int)((n4 + 255) / 256), 256, 0, stream>>>((float4*)agg, n4);
      long long th = (long long)E2 * 16;
      scatter_add_kernel<<<(int)((th + 255) / 256), 256, 0, stream>>>(xt_in, e_t2s, agg, E2);
      sage_gemm_kernel<<<(NS + 127) / 128, 256, 0, stream>>>(
          agg, xs_in, Wl_t2s + (size_t)l * 4096, Wr_t2s + (size_t)l * 4096,
          b_t2s + (size_t)l * 64, xs_out, NS);
    }
    xs_in = xs_out;
    xt_in = xt_out;
  }

  // ---- global add pool over both node sets, then final linear
  fill_zero_kernel<<<1, 32, 0, stream>>>((float4*)pooled, 16);
  pool_kernel<<<512, 256, 0, stream>>>(xs_in, pooled, NS);
  pool_kernel<<<512, 256, 0, stream>>>(xt_in, pooled, NT);
  finish_kernel<<<1, 64, 0, stream>>>(pooled, W_lin, b_lin, out);
}